// NuclearNormRegularizer_18983755448635
// MI455X (gfx1250) — compile-verified
//
#include <hip/hip_runtime.h>
#include <hip/hip_bf16.h>

// CDNA5 / gfx1250: wave32 WMMA f32 16x16x4. One wave per block computes a
// 32x32 output tile of D = L^T * R (L read column-sliced == L when L is
// symmetric), with K stepping by 4 per WMMA.

typedef __attribute__((ext_vector_type(2))) float v2f;
typedef __attribute__((ext_vector_type(8))) float v8f;

#define LDC 256  // all matrices here have 256 columns

__device__ __forceinline__ void store8(float* __restrict__ D, int mb, int n,
                                       v8f c, float alpha, float diag) {
#pragma unroll
  for (int v = 0; v < 8; ++v) {
    float val = alpha * c[v];
    if (mb + v == n) val += diag;     // branchless on gfx1250 (v_cmp+cndmask)
    D[(size_t)(mb + v) * LDC + n] = val;
  }
}

// D[i][j] = alpha * sum_k L[k][i] * R[k][j]  (+ diag on i==j)
// grid = (64, nslots); block = 32 (one wave). Tile t: i0=(t>>3)*32, j0=(t&7)*32.
__global__ __launch_bounds__(32) void gemm_symT(
    const float* __restrict__ Lb, const float* __restrict__ Rb,
    float* __restrict__ Db, long strideL, long strideR, long strideD,
    int K, float alpha, float diag) {
  const int lane = threadIdx.x;                 // 0..31
  const int t    = blockIdx.x;                  // 0..63
  const int slot = blockIdx.y;
  const float* L = Lb + (size_t)slot * strideL;
  const float* R = Rb + (size_t)slot * strideR;
  float*       D = Db + (size_t)slot * strideD;

  const int i0 = (t >> 3) << 5;
  const int j0 = (t & 7) << 5;
  const int khalf = (lane >> 4) << 1;           // lanes 0-15 -> K=0,1 ; 16-31 -> K=2,3
  const int cl    = lane & 15;

  v8f c00 = {}, c01 = {}, c10 = {}, c11 = {};

  for (int k = 0; k < K; k += 4) {
    const float* Lr = L + (size_t)(k + khalf) * LDC;
    const float* Rr = R + (size_t)(k + khalf) * LDC;
    v2f a0, a1, b0, b1;
    a0.x = Lr[i0 + cl];        a0.y = Lr[LDC + i0 + cl];
    a1.x = Lr[i0 + 16 + cl];   a1.y = Lr[LDC + i0 + 16 + cl];
    b0.x = Rr[j0 + cl];        b0.y = Rr[LDC + j0 + cl];
    b1.x = Rr[j0 + 16 + cl];   b1.y = Rr[LDC + j0 + 16 + cl];
    c00 = __builtin_amdgcn_wmma_f32_16x16x4_f32(false, a0, false, b0, (short)0, c00, false, false);
    c01 = __builtin_amdgcn_wmma_f32_16x16x4_f32(false, a0, false, b1, (short)0, c01, false, false);
    c10 = __builtin_amdgcn_wmma_f32_16x16x4_f32(false, a1, false, b0, (short)0, c10, false, false);
    c11 = __builtin_amdgcn_wmma_f32_16x16x4_f32(false, a1, false, b1, (short)0, c11, false, false);
  }

  const int n0  = j0 + cl;
  const int n1  = j0 + 16 + cl;
  const int mb0 = i0 + ((lane >> 4) << 3);      // +0 or +8 per C/D layout
  const int mb1 = mb0 + 16;
  store8(D, mb0, n0, c00, alpha, diag);
  store8(D, mb0, n1, c01, alpha, diag);
  store8(D, mb1, n0, c10, alpha, diag);
  store8(D, mb1, n1, c11, alpha, diag);
}

// Per bond (1 block, 256 threads): fro2 = ||G||_F^2; Y *= 1/||G||_F; Z = I;
// sf[b] = sqrt(||G||_F)  (since trace(sqrt(G)) = sqrt(||G||_F)*trace(sqrt(A))).
__global__ __launch_bounds__(256) void prep_kernel(
    float* __restrict__ mats, long slotStride, long yOff, long zOff,
    float* __restrict__ sf_arr) {
  const int slot = blockIdx.x;
  float* Y = mats + (size_t)slot * slotStride + yOff;
  float* Z = mats + (size_t)slot * slotStride + zOff;
  const int t = threadIdx.x;
  __shared__ float red[256];
  float s = 0.f;
  for (int j = 0; j < 256; ++j) {
    float v = Y[(size_t)j * 256 + t];           // coalesced across threads
    s += v * v;
  }
  red[t] = s;
  __syncthreads();
  for (int w = 128; w > 0; w >>= 1) {
    if (t < w) red[t] += red[t + w];
    __syncthreads();
  }
  const float fro2 = red[0];
  const float invf = rsqrtf(fro2);              // 1/||G||_F
  if (t == 0) sf_arr[slot] = sqrtf(sqrtf(fro2));
  for (int j = 0; j < 256; ++j) {
    const int idx = j * 256 + t;
    Y[idx] *= invf;
    Z[idx] = ((idx >> 8) == (idx & 255)) ? 1.0f : 0.0f;
  }
}

__global__ __launch_bounds__(256) void trace_kernel(
    const float* __restrict__ matsY, long slotStride,
    const float* __restrict__ sf_arr, float* __restrict__ bond_res) {
  const int slot = blockIdx.x;
  const float* Y = matsY + (size_t)slot * slotStride;
  const int t = threadIdx.x;
  __shared__ float red[256];
  red[t] = Y[(size_t)t * 257];                  // diagonal
  __syncthreads();
  for (int w = 128; w > 0; w >>= 1) {
    if (t < w) red[t] += red[t + w];
    __syncthreads();
  }
  if (t == 0) bond_res[slot] = sf_arr[slot] * red[0];
}

__global__ void finalize_kernel(const float* __restrict__ bond_res, int nb,
                                float* __restrict__ out) {
  if (threadIdx.x == 0 && blockIdx.x == 0) {
    float s = 0.f;
    for (int i = 0; i < nb; ++i) s += bond_res[i];   // fixed order: deterministic
    out[0] = 1e-4f * (s / (float)nb);
  }
}

extern "C" void kernel_launch(void* const* d_in, const int* in_sizes, int n_in,
                              void* d_out, int out_size, void* d_ws, size_t ws_size,
                              hipStream_t stream) {
  (void)in_sizes; (void)n_in; (void)out_size;
  const float* cores = (const float*)d_in[0];

  constexpr int  R = 256, Dk = 4, ROWS = R * Dk;   // 1024 x 256 unfolding
  constexpr int  NB = 127;                         // bonds
  constexpr long MATF  = 256L * 256L;              // floats per 256x256 matrix
  constexpr long SLOTF = 4 * MATF;                 // 4 rotating buffers per bond
  constexpr size_t SLOTB = (size_t)SLOTF * sizeof(float);
  constexpr long CORF = (long)ROWS * R;            // floats per core
  const int ITERS = 18;

  char*  ws       = (char*)d_ws;
  float* bond_res = (float*)ws;                    // 128 floats
  float* sf_arr   = (float*)(ws + 512);            // 128 floats
  float* mats     = (float*)(ws + 4096);

  int slots = 1;
  if (ws_size > 4096) {
    size_t s = (ws_size - 4096) / SLOTB;
    slots = (s < 1) ? 1 : (s > (size_t)NB ? NB : (int)s);
  }

  for (int b0 = 0; b0 < NB; b0 += slots) {
    const int cnt = (NB - b0 < slots) ? (NB - b0) : slots;
    int rY = 0, rZ = 1, rW = 2, rT = 3;
    const float* C0 = cores + (size_t)b0 * CORF;

    // Gram: Y = M^T M  (K = 1024)
    gemm_symT<<<dim3(64, cnt), 32, 0, stream>>>(
        C0, C0, mats + rY * MATF, CORF, CORF, SLOTF, ROWS, 1.0f, 0.0f);

    // Normalize Y, set Z = I, record sqrt(||G||_F)
    prep_kernel<<<cnt, 256, 0, stream>>>(mats, SLOTF, rY * MATF, rZ * MATF,
                                         sf_arr + b0);

    // Coupled Newton-Schulz
    for (int it = 0; it < ITERS; ++it) {
      // W = 1.5 I - 0.5 * Z*Y
      gemm_symT<<<dim3(64, cnt), 32, 0, stream>>>(
          mats + rZ * MATF, mats + rY * MATF, mats + rW * MATF,
          SLOTF, SLOTF, SLOTF, 256, -0.5f, 1.5f);
      // T = Y*W
      gemm_symT<<<dim3(64, cnt), 32, 0, stream>>>(
          mats + rY * MATF, mats + rW * MATF, mats + rT * MATF,
          SLOTF, SLOTF, SLOTF, 256, 1.0f, 0.0f);
      // Z' = W*Z  -> old Y slot (Y no longer needed)
      gemm_symT<<<dim3(64, cnt), 32, 0, stream>>>(
          mats + rW * MATF, mats + rZ * MATF, mats + rY * MATF,
          SLOTF, SLOTF, SLOTF, 256, 1.0f, 0.0f);
      const int nY = rT, nZ = rY, nW = rZ, nT = rW;  // cyclic role rotation
      rY = nY; rZ = nZ; rW = nW; rT = nT;
    }

    // nuclear_b = sqrt(||G||_F) * trace(Y_final)
    trace_kernel<<<cnt, 256, 0, stream>>>(mats + rY * MATF, SLOTF,
                                          sf_arr + b0, bond_res + b0);
  }

  finalize_kernel<<<1, 1, 0, stream>>>(bond_res, NB, (float*)d_out);
}